// FasterRcnnFeatureExtractor_79594333929872
// MI455X (gfx1250) — compile-verified
//
#include <hip/hip_runtime.h>
#include <hip/hip_bf16.h>

#define NUM_CLASSES 91
#define FG_CLASSES  90
#define KEEP        100
#define PROPS_PER_BLK 128
#define CAND_PER_BLK  100
#define BBOX_CLIP 4.135166556742356f   // log(1000/16)
#define NEG_INF (-__builtin_inff())

typedef float v2f __attribute__((ext_vector_type(2)));
typedef float v8f __attribute__((ext_vector_type(8)));
typedef unsigned long long u64;
typedef unsigned int u32;

__device__ __forceinline__ u64 pack_key(float s, u32 gidx) {
  // score in high 32 bits (softmax scores are >=0 so float bits order == value order);
  // ~gidx in low bits so ties prefer the smaller index (stable argsort order).
  return ((u64)__float_as_uint(s) << 32) | (u64)(~gidx);
}

// In-LDS bitonic sort, descending, N elements, T threads.
template<int N, int T>
__device__ __forceinline__ void bitonic_desc(u64* keys, int tid) {
  for (unsigned k = 2; k <= (unsigned)N; k <<= 1) {
    for (unsigned j = k >> 1; j > 0; j >>= 1) {
      for (unsigned i = (unsigned)tid; i < (unsigned)N; i += (unsigned)T) {
        unsigned ixj = i ^ j;
        if (ixj > i) {
          u64 a = keys[i], b = keys[ixj];
          bool up = ((i & k) == 0);            // descending region
          bool sw = up ? (a < b) : (a > b);
          if (sw) { keys[i] = b; keys[ixj] = a; }
        }
      }
      __syncthreads();
    }
  }
}

// ---------------------------------------------------------------------------
// Pass A: softmax + per-block top-100 candidate selection.
// Wave layout: 16 proposals per wave; lane L (pl=L%16, h=L/16) owns classes
// {4j+2h, 4j+2h+1} of proposal pl — exactly the V_WMMA_F32_16X16X4_F32 A-matrix
// layout, so the cross-lane sum-of-exp reduction is one WMMA vs an all-ones B.
// ---------------------------------------------------------------------------
__global__ __launch_bounds__(256) void rcnn_score_topk(
    const float* __restrict__ logits, u64* __restrict__ cand,
    int P, int blocksPerImg)
{
  __shared__ u64 keys[2048];
  const int tid  = threadIdx.x;
  const int img  = blockIdx.x / blocksPerImg;
  const int blk  = blockIdx.x % blocksPerImg;
  const int lane = tid & 31;
  const int wave = tid >> 5;
  const int h    = lane >> 4;
  const int pl   = lane & 15;
  const int prop = blk * PROPS_PER_BLK + wave * 16 + pl;    // image-local
  const long long row = (long long)(img * P + prop) * NUM_CLASSES;

  // Load this lane's 46 logits; track max.
  float vals[23][2];
  float mx = NEG_INF;
  #pragma unroll
  for (int j = 0; j < 23; ++j) {
    int c0 = 4 * j + 2 * h;                                  // c0 <= 90 always
    float a = logits[row + c0];
    float b = (c0 + 1 < NUM_CLASSES) ? logits[row + c0 + 1] : NEG_INF;
    vals[j][0] = a; vals[j][1] = b;
    mx = fmaxf(mx, fmaxf(a, b));
  }
  // Proposal max = combine the two half-lanes.
  mx = fmaxf(mx, __shfl_xor(mx, 16, 32));

  // exp + per-lane partial sum (exp(-inf)=0 handles the padded slot).
  float psum = 0.f;
  #pragma unroll
  for (int j = 0; j < 23; ++j) {
    float e0 = expf(vals[j][0] - mx);
    float e1 = expf(vals[j][1] - mx);
    vals[j][0] = e0; vals[j][1] = e1;
    psum += e0 + e1;
  }

  // Cross-lane reduction on the matrix pipe:
  // A[pl, 2h] = psum, A[pl, 2h+1] = 0, B = ones => D[pl, n] = rowsum(pl) for all n.
  v2f a2; a2[0] = psum; a2[1] = 0.f;
  v2f b2; b2[0] = 1.f;  b2[1] = 1.f;
  v8f acc = {};
  acc = __builtin_amdgcn_wmma_f32_16x16x4_f32(
      /*neg_a=*/false, a2, /*neg_b=*/false, b2,
      /*c_mod=*/(short)0, acc, /*reuse_a=*/false, /*reuse_b=*/false);

  // D layout: vgpr r = row M=r (lanes 0-15) / M=r+8 (lanes 16-31), replicated cols.
  int r = pl & 7;
  float t = acc[0];
  #pragma unroll
  for (int i = 1; i < 8; ++i) t = (r == i) ? acc[i] : t;
  float to = __shfl_xor(t, 16, 32);
  float total = (((pl < 8)) == (lane < 16)) ? t : to;
  float inv = 1.0f / total;

  // Per-lane top-8 (sorted descending) over this lane's foreground scores.
  u64 top[8] = {0ull,0ull,0ull,0ull,0ull,0ull,0ull,0ull};
  #pragma unroll
  for (int j = 0; j < 23; ++j) {
    #pragma unroll
    for (int k = 0; k < 2; ++k) {
      int c = 4 * j + 2 * h + k;
      if (c < 1 || c >= NUM_CLASSES) continue;               // skip background/pad
      float sc = vals[j][k] * inv;
      u32 gidx = (u32)prop * (u32)FG_CLASSES + (u32)(c - 1); // p*90 + (c-1)
      u64 key = pack_key(sc, gidx);
      if (key > top[7]) {
        top[7] = key;
        #pragma unroll
        for (int i = 7; i > 0; --i) {
          if (top[i] > top[i-1]) { u64 tmp = top[i-1]; top[i-1] = top[i]; top[i] = tmp; }
        }
      }
    }
  }

  #pragma unroll
  for (int i = 0; i < 8; ++i) keys[tid * 8 + i] = top[i];
  __syncthreads();
  bitonic_desc<2048, 256>(keys, tid);

  if (tid < CAND_PER_BLK)
    cand[((long long)img * blocksPerImg + blk) * CAND_PER_BLK + tid] = keys[tid];
}

// ---------------------------------------------------------------------------
// Pass B: per-image merge of candidates, then decode/clip/scale the 100 kept.
// ---------------------------------------------------------------------------
__global__ __launch_bounds__(512) void rcnn_merge_decode(
    const u64* __restrict__ cand, const float* __restrict__ boxreg,
    const float* __restrict__ props, float* __restrict__ outBoxes,
    u32* __restrict__ featIdx, const int* __restrict__ imgH,
    const int* __restrict__ imgW, int P, int candPerImg)
{
  __shared__ u64 keys[4096];
  const int tid = threadIdx.x;
  const int img = blockIdx.x;
  for (int i = tid; i < 4096; i += 512)
    keys[i] = (i < candPerImg) ? cand[(long long)img * candPerImg + i] : 0ull;
  __syncthreads();
  bitonic_desc<4096, 512>(keys, tid);

  if (tid < KEEP) {
    u64 key  = keys[tid];
    u32 gidx = ~((u32)(key & 0xffffffffull));
    int p = (int)(gidx / (u32)FG_CLASSES);
    int c = (int)(gidx % (u32)FG_CLASSES) + 1;               // real class (bg dropped)
    long long rowN = (long long)img * P + p;

    float x1 = props[rowN*4+0], y1 = props[rowN*4+1];
    float x2 = props[rowN*4+2], y2 = props[rowN*4+3];
    float w  = x2 - x1, hh = y2 - y1;
    float cx = x1 + 0.5f * w, cy = y1 + 0.5f * hh;

    const float* rg = boxreg + rowN * (4 * NUM_CLASSES) + 4 * c;
    float dx = rg[0] / 10.0f;
    float dy = rg[1] / 10.0f;
    float dw = fminf(rg[2] / 5.0f, BBOX_CLIP);
    float dh = fminf(rg[3] / 5.0f, BBOX_CLIP);

    float pcx = dx * w + cx, pcy = dy * hh + cy;
    float pw  = expf(dw) * w, ph = expf(dh) * hh;
    float bx1 = pcx - 0.5f * pw, by1 = pcy - 0.5f * ph;
    float bx2 = pcx + 0.5f * pw, by2 = pcy + 0.5f * ph;

    float W = (float)imgW[0], H = (float)imgH[0];
    bx1 = fminf(fmaxf(bx1, 0.f), W); bx2 = fminf(fmaxf(bx2, 0.f), W);
    by1 = fminf(fmaxf(by1, 0.f), H); by2 = fminf(fmaxf(by2, 0.f), H);

    float* o = outBoxes + ((long long)img * KEEP + tid) * 4;
    o[0] = bx1 / H; o[1] = by1 / H; o[2] = bx2 / H; o[3] = by2 / H;  // ref divides by image_h
    featIdx[img * KEEP + tid] = (u32)p;
  }
}

// ---------------------------------------------------------------------------
// Pass C: feature gather through LDS using CDNA5 async global<->LDS ops.
// One block per kept detection; each lane moves 16B via the ASYNCcnt path.
// ---------------------------------------------------------------------------
__global__ __launch_bounds__(256) void rcnn_gather_feats(
    const float* __restrict__ feats, const u32* __restrict__ featIdx,
    float* __restrict__ outFeats, int P, int FD)
{
  __shared__ float4 buf[256];
  const int tid = threadIdx.x;
  const int img = blockIdx.x / KEEP;
  const int d   = blockIdx.x % KEEP;
  const u32 p   = featIdx[img * KEEP + d];
  const float4* src = (const float4*)(feats + ((long long)img * P + (long long)p) * FD);
  float4*       dst = (float4*)(outFeats + ((long long)img * KEEP + d) * FD);
  const int nvec = FD >> 2;

  unsigned lds = (unsigned)(unsigned long long)(uintptr_t)&buf[tid];
  for (int i = tid; i < nvec; i += 256) {
    unsigned long long ga = (unsigned long long)(uintptr_t)(src + i);
    unsigned long long gd = (unsigned long long)(uintptr_t)(dst + i);
    asm volatile("global_load_async_to_lds_b128 %0, %1, off"
                 :: "v"(lds), "v"(ga) : "memory");
    asm volatile("s_wait_asynccnt 0" ::: "memory");
    asm volatile("global_store_async_from_lds_b128 %0, %1, off"
                 :: "v"(gd), "v"(lds) : "memory");
    asm volatile("s_wait_asynccnt 0" ::: "memory");
  }
}

// ---------------------------------------------------------------------------
extern "C" void kernel_launch(void* const* d_in, const int* in_sizes, int n_in,
                              void* d_out, int out_size, void* d_ws, size_t ws_size,
                              hipStream_t stream) {
  const float* logits = (const float*)d_in[0];
  const float* boxreg = (const float*)d_in[1];
  const float* props  = (const float*)d_in[2];
  const float* feats  = (const float*)d_in[3];
  const int*   imgH   = (const int*)d_in[5];
  const int*   imgW   = (const int*)d_in[6];

  const int N  = in_sizes[0] / NUM_CLASSES;        // 32768
  const int FD = in_sizes[3] / N;                  // 1024
  const int B  = out_size / (KEEP * (4 + FD));     // 8
  const int P  = N / B;                            // 4096
  const int blocksPerImg = P / PROPS_PER_BLK;      // 32
  const int candPerImg   = blocksPerImg * CAND_PER_BLK;  // 3200 (<= 4096)

  u64* cand    = (u64*)d_ws;
  u32* featIdx = (u32*)((char*)d_ws + (size_t)B * candPerImg * sizeof(u64));
  float* outBoxes = (float*)d_out;
  float* outFeats = (float*)d_out + (size_t)B * KEEP * 4;

  rcnn_score_topk<<<B * blocksPerImg, 256, 0, stream>>>(logits, cand, P, blocksPerImg);
  rcnn_merge_decode<<<B, 512, 0, stream>>>(cand, boxreg, props, outBoxes,
                                           featIdx, imgH, imgW, P, candPerImg);
  rcnn_gather_feats<<<B * KEEP, 256, 0, stream>>>(feats, featIdx, outFeats, P, FD);
}